// PairingLayer_75711683493997
// MI455X (gfx1250) — compile-verified
//
#include <hip/hip_runtime.h>

typedef __attribute__((ext_vector_type(2))) float v2f;
typedef __attribute__((ext_vector_type(4))) float v4f;
typedef __attribute__((ext_vector_type(8))) float v8f;

#define B_ 8
#define N_ 128
#define D_ 768
#define KCHUNK 32
#define LDS_STRIDE 36   // 32 + 4 pad: keeps float4 alignment, spreads LDS banks

// ---------------------------------------------------------------------------
// Kernel 1: per-row dot products si = X·w1, sj = X·w2  (one wave32 per row)
// ---------------------------------------------------------------------------
__global__ void sisj_kernel(const float* __restrict__ X,
                            const float* __restrict__ W,
                            float* __restrict__ ws_si,
                            float* __restrict__ ws_sj) {
  int tid  = threadIdx.x;
  int lane = tid & 31;
  int g    = blockIdx.x * (blockDim.x >> 5) + (tid >> 5);  // row id 0..B*N-1
  const float* row = X + (size_t)g * D_;
  const float* w1  = W;
  const float* w2  = W + D_;

  float s1 = 0.f, s2 = 0.f;
#pragma unroll
  for (int t = 0; t < D_ / 32; ++t) {
    int c = lane + t * 32;
    float x = row[c];
    s1 += x * w1[c];
    s2 += x * w2[c];
  }
#pragma unroll
  for (int off = 16; off > 0; off >>= 1) {
    s1 += __shfl_down(s1, off, 32);
    s2 += __shfl_down(s2, off, 32);
  }
  if (lane == 0) { ws_si[g] = s1; ws_sj[g] = s2; }
}

// ---------------------------------------------------------------------------
// Kernel 2: scores[b,i,j] = (X_b*wm)@X_b^T + si[i] + sj[j] + wr[j-i+N-1] + b0,
//           masked to -inf. 256 threads = 8 wave32; each wave one 16x16 tile
//           via V_WMMA_F32_16X16X4_F32, K chunked by 32 through LDS.
// ---------------------------------------------------------------------------
__global__ void pair_kernel(const float* __restrict__ X,
                            const float* __restrict__ W,
                            const float* __restrict__ bptr,
                            const int*   __restrict__ mask,
                            const float* __restrict__ ws_si,
                            const float* __restrict__ ws_sj,
                            float* __restrict__ out) {
  __shared__ float ldsA[32 * LDS_STRIDE];  // wm-scaled A panel: 32 rows x KCHUNK
  __shared__ float ldsB[64 * LDS_STRIDE];  // B panel: 64 rows x KCHUNK

  int bid   = blockIdx.x;
  int b     = bid >> 3;
  int rem   = bid & 7;
  int mBase = (rem >> 1) * 32;   // 0,32,64,96
  int nBase = (rem & 1) * 64;    // 0,64

  int tid   = threadIdx.x;
  int lane  = tid & 31;
  int wv    = tid >> 5;          // wave 0..7
  int mTile = wv >> 2;           // 0..1 : which 16-row strip of the 32
  int nTile = wv & 3;            // 0..3 : which 16-col strip of the 64

  const float* Xb = X + (size_t)b * N_ * D_;
  const float* wm = W + 2 * D_;
  const float* wr = W + 3 * D_;

  v8f acc = {};

  // staging assignments
  int arow = tid >> 3;          // 0..31 (A row)
  int ac4  = (tid & 7) * 4;     // 0..28 (float4 column)

  int frow = lane & 15;         // fragment row (M for A, N for B)
  int koff = (lane >> 4) * 2;   // lanes 16-31 hold K+2,K+3 per ISA layout
  const float* aPtr = &ldsA[(mTile * 16 + frow) * LDS_STRIDE + koff];
  const float* bPtr = &ldsB[(nTile * 16 + frow) * LDS_STRIDE + koff];

  for (int k0 = 0; k0 < D_; k0 += KCHUNK) {
    // ---- stage A panel (scaled by wm), one float4 per thread ----
    {
      v4f x = *(const v4f*)(Xb + (size_t)(mBase + arow) * D_ + k0 + ac4);
      v4f s = *(const v4f*)(wm + k0 + ac4);
      x *= s;
      *(v4f*)(&ldsA[arow * LDS_STRIDE + ac4]) = x;
    }
    // ---- stage B panel, two float4 per thread ----
#pragma unroll
    for (int it = 0; it < 2; ++it) {
      int idx = tid + it * 256;
      int br  = idx >> 3;          // 0..63
      int bc4 = (idx & 7) * 4;
      v4f x = *(const v4f*)(Xb + (size_t)(nBase + br) * D_ + k0 + bc4);
      *(v4f*)(&ldsB[br * LDS_STRIDE + bc4]) = x;
    }
    __syncthreads();

    // ---- 8 x V_WMMA_F32_16X16X4_F32 over this K chunk ----
#pragma unroll
    for (int step = 0; step < KCHUNK / 4; ++step) {
      v2f af = *(const v2f*)(aPtr + step * 4);
      v2f bf = *(const v2f*)(bPtr + step * 4);
      acc = __builtin_amdgcn_wmma_f32_16x16x4_f32(
          /*neg_a=*/false, af, /*neg_b=*/false, bf,
          /*c_mod=*/(short)0, acc, /*reuse_a=*/false, /*reuse_b=*/false);
    }
    __syncthreads();
  }

  // ---- epilogue: C layout is VGPR r -> M = r + 8*(lane>=16), N = lane&15 ----
  int j     = nBase + nTile * 16 + (lane & 15);
  int iBase = mBase + mTile * 16 + (lane >> 4) * 8;
  float b0  = bptr[0];
  float sjv = ws_sj[b * N_ + j];
  int   mj  = mask[b * N_ + j];

#pragma unroll
  for (int r = 0; r < 8; ++r) {
    int   i   = iBase + r;
    int   mi  = mask[b * N_ + i];
    float val = acc[r] + ws_si[b * N_ + i] + sjv + wr[j - i + N_ - 1] + b0;
    bool  ok  = (mi != 0) && (mj != 0);
    out[(size_t)(b * N_ + i) * N_ + j] = ok ? val : -__builtin_inff();
  }
}

// ---------------------------------------------------------------------------
extern "C" void kernel_launch(void* const* d_in, const int* in_sizes, int n_in,
                              void* d_out, int out_size, void* d_ws, size_t ws_size,
                              hipStream_t stream) {
  const float* X    = (const float*)d_in[0];  // (B,N,D) fp32
  const float* W    = (const float*)d_in[1];  // (F,)    fp32
  const float* bp   = (const float*)d_in[2];  // (1,)    fp32
  const int*   msk  = (const int*)d_in[3];    // (B,N)   int32
  float*       out  = (float*)d_out;          // (B,N,N) fp32

  float* ws_si = (float*)d_ws;                // B*N floats
  float* ws_sj = ws_si + B_ * N_;             // B*N floats

  // 1024 rows, 8 waves per 256-thread block -> 128 blocks
  sisj_kernel<<<dim3(B_ * N_ / 8), dim3(256), 0, stream>>>(X, W, ws_si, ws_sj);

  // 8 batches x (4 m-blocks) x (2 n-blocks) = 64 blocks of 256 threads
  pair_kernel<<<dim3(64), dim3(256), 0, stream>>>(X, W, bp, msk, ws_si, ws_sj, out);
}